// PHD_conv2d_24885040513059
// MI455X (gfx1250) — compile-verified
//
#include <hip/hip_runtime.h>
#include <hip/hip_bf16.h>

#define W_DIM  81920
#define K_TAPS 10
#define CIN    64
#define COUT   64
#define NB     4

#define M_TILE 32
#define GS_STRIDE 68   // padded row stride (floats): conflict-free float2 LDS reads, 16B-aligned rows
#define BS_STRIDE 68
#define CS_STRIDE 36   // multiple of 4 so epilogue float4 reads stay 16B-aligned

typedef __attribute__((ext_vector_type(2))) float v2f;
typedef __attribute__((ext_vector_type(8))) float v8f;
typedef __attribute__((ext_vector_type(4))) int   v4i;

// ---- CDNA5 async global->LDS path (guarded; falls back to sync copy) ----
#if defined(__HIP_DEVICE_COMPILE__) && \
    __has_builtin(__builtin_amdgcn_global_load_async_to_lds_b128) && \
    __has_builtin(__builtin_amdgcn_s_wait_asynccnt)
#define USE_ASYNC 1
#else
#define USE_ASYNC 0
#endif

#if USE_ASYNC
#define ASYNC_CP_F4(gsrc, ldst)                                              \
    __builtin_amdgcn_global_load_async_to_lds_b128(                          \
        (__attribute__((address_space(1))) v4i*)(gsrc),                      \
        (__attribute__((address_space(3))) v4i*)(ldst), 0, 0)
#define WAIT_ASYNC(n) __builtin_amdgcn_s_wait_asynccnt(n)
#else
#define ASYNC_CP_F4(gsrc, ldst) (*(float4*)(ldst) = *(const float4*)(gsrc))
#define WAIT_ASYNC(n)
#endif

// ---------------- Kernel 0: pack weight (o,c,k) -> Wk[k][o][c] ----------------
__global__ __launch_bounds__(256) void PHD_pack_weight(const float* __restrict__ wsrc,
                                                       float* __restrict__ wk) {
    int g = blockIdx.x * 256 + threadIdx.x;
    if (g >= COUT * CIN * K_TAPS) return;
    int k   = g / (COUT * CIN);
    int rem = g - k * (COUT * CIN);
    int o   = rem >> 6;
    int c   = rem & 63;
    wk[g] = wsrc[(o * CIN + c) * K_TAPS + k];   // weight flat: o*640 + c*10 + k
}

// ---------------- Kernel 1: transpose x (n,c,w) -> xt[(n,w),c] ----------------
__global__ __launch_bounds__(256) void PHD_transpose(const float* __restrict__ x,
                                                     float* __restrict__ xt) {
    __shared__ float tile[64 * 65];
    const int t  = threadIdx.x;
    const int b  = blockIdx.x;                 // 0 .. NB*(W/64)-1
    const int n  = b / (W_DIM / 64);
    const int w0 = (b - n * (W_DIM / 64)) * 64;
    const float* xn = x + (size_t)n * CIN * W_DIM;
    #pragma unroll
    for (int i = 0; i < 16; ++i) {             // coalesced along w
        int linear = t + i * 256;
        int c = linear >> 6;
        int w = linear & 63;
        tile[c * 65 + w] = xn[(size_t)c * W_DIM + (w0 + w)];
    }
    __syncthreads();
    float* xtn = xt + ((size_t)n * W_DIM + w0) * CIN;
    #pragma unroll
    for (int i = 0; i < 16; ++i) {             // coalesced along c
        int linear = t + i * 256;
        int w = linear >> 6;
        int c = linear & 63;
        xtn[(size_t)w * CIN + c] = tile[c * 65 + w];
    }
}

// ---------------- Kernel 2: gather + WMMA GEMM + bias + ReLU ----------------
// Block: 256 threads = 8 wave32s arranged 2(M) x 4(N); tile = 32 rows x 64 outs.
// Double-buffered async LDS staging; 6 x b128 copies per thread per tap (2 gather + 4 weight).
__global__ __launch_bounds__(256) void PHD_conv_gemm(const float* __restrict__ xt,
                                                     const float* __restrict__ wk,
                                                     const int*   __restrict__ table,
                                                     const float* __restrict__ bias,
                                                     float*       __restrict__ out) {
    __shared__ __align__(16) float Gs[2][M_TILE * GS_STRIDE];  // gathered A: 32 rows x 64 ch
    __shared__ __align__(16) float Bs[2][COUT * BS_STRIDE];    // B^T: Bs[o][c]
    __shared__ __align__(16) float Cs[COUT * CS_STRIDE];       // epilogue: Cs[o][w_local]

    const int t     = threadIdx.x;
    const int pBase = blockIdx.x * M_TILE;     // 32 | W  => block never straddles n
    const int n     = pBase / W_DIM;
    const int w0    = pBase - n * W_DIM;

    const int lane  = t & 31;
    const int wv    = t >> 5;
    const int mtile = wv >> 2;                 // 0..1 : which 16-row half
    const int ob    = (wv & 3) << 4;           // 0,16,32,48 : output-channel base
    const int m     = lane & 15;
    const int half  = lane >> 4;

    v8f acc = {};

    // ---- stage tap k into buffer buf: 2 gather-f4 + 4 weight-f4 per thread ----
    auto stage = [&](int k, int buf) {
        float* GsB = Gs[buf];
        float* BsB = Bs[buf];
        #pragma unroll
        for (int i = 0; i < 2; ++i) {          // 512 float4 slots: 32 rows x 16
            int s  = t + i * 256;
            int r  = s >> 4;
            int c4 = (s & 15) << 2;
            int idx = table[(w0 + r) * K_TAPS + k];
            ASYNC_CP_F4(&xt[((size_t)n * W_DIM + idx) * CIN + c4],
                        &GsB[r * GS_STRIDE + c4]);
        }
        const float* wkk = wk + k * (COUT * CIN);
        #pragma unroll
        for (int i = 0; i < 4; ++i) {          // 1024 float4 slots: 64 o x 16
            int s  = t + i * 256;
            int o  = s >> 4;
            int c4 = (s & 15) << 2;
            ASYNC_CP_F4(&wkk[(o << 6) + c4], &BsB[o * BS_STRIDE + c4]);
        }
    };

    stage(0, 0);
    for (int k = 0; k < K_TAPS; ++k) {
        const int b = k & 1;
        if (k + 1 < K_TAPS) {
            stage(k + 1, b ^ 1);               // buf b^1 last read in iter k-1 (barrier'd)
            WAIT_ASYNC(6);                     // stage k done; k+1's 6 copies stay in flight
        } else {
            WAIT_ASYNC(0);
        }
        __syncthreads();
        const float* GsB = Gs[b];
        const float* BsB = Bs[b];
        #pragma unroll
        for (int c0 = 0; c0 < 64; c0 += 4) {   // 16 WMMA steps over 64-wide K chunk
            int col = c0 + half * 2;
            // A lane layout: VGPR0/1 = (M=m, K=col/col+1)   -> contiguous float2
            v2f a = *(const v2f*)&GsB[(mtile * 16 + m) * GS_STRIDE + col];
            // B lane layout: VGPR0/1 = (K=col/col+1, N=ob+m) -> contiguous float2 of B^T
            v2f bb = *(const v2f*)&BsB[(ob + m) * BS_STRIDE + col];
            acc = __builtin_amdgcn_wmma_f32_16x16x4_f32(
                /*neg_a=*/false, a, /*neg_b=*/false, bb,
                /*c_mod=*/(short)0, acc, /*reuse_a=*/false, /*reuse_b=*/false);
        }
        __syncthreads();                       // all reads of buf b done before restage
    }

    // ---- stage C through LDS so global stores are wide contiguous runs ----
    #pragma unroll
    for (int j = 0; j < 8; ++j) {
        int row = mtile * 16 + half * 8 + j;   // C layout: VGPR j -> M = j + 8*half
        Cs[(ob + m) * CS_STRIDE + row] = acc[j];
    }
    __syncthreads();
    float* outn = out + (size_t)n * COUT * W_DIM;
    #pragma unroll
    for (int i = 0; i < 2; ++i) {              // 512 float4 slots: 64 o x 8
        int s  = t + i * 256;
        int o  = s >> 3;
        int w4 = (s & 7) << 2;
        float bv = bias[o];
        float4 v;
        v.x = Cs[o * CS_STRIDE + w4 + 0] + bv;
        v.y = Cs[o * CS_STRIDE + w4 + 1] + bv;
        v.z = Cs[o * CS_STRIDE + w4 + 2] + bv;
        v.w = Cs[o * CS_STRIDE + w4 + 3] + bv;
        v.x = v.x > 0.0f ? v.x : 0.0f;
        v.y = v.y > 0.0f ? v.y : 0.0f;
        v.z = v.z > 0.0f ? v.z : 0.0f;
        v.w = v.w > 0.0f ? v.w : 0.0f;
        *(float4*)&outn[(size_t)o * W_DIM + (w0 + w4)] = v;
    }
}

extern "C" void kernel_launch(void* const* d_in, const int* in_sizes, int n_in,
                              void* d_out, int out_size, void* d_ws, size_t ws_size,
                              hipStream_t stream) {
    const float* x      = (const float*)d_in[0];
    const int*   table  = (const int*)  d_in[1];
    const float* weight = (const float*)d_in[2];
    const float* bias   = (const float*)d_in[3];
    float* out = (float*)d_out;

    // workspace: [ xt : NB*W*CIN f32 = 83,886,080 B ][ Wk : 163,840 B ]
    float* xt = (float*)d_ws;
    float* wk = (float*)((char*)d_ws + (size_t)NB * W_DIM * CIN * sizeof(float));

    PHD_transpose  <<<NB * (W_DIM / 64), 256, 0, stream>>>(x, xt);
    PHD_pack_weight<<<(COUT * CIN * K_TAPS + 255) / 256, 256, 0, stream>>>(weight, wk);
    PHD_conv_gemm  <<<(NB * W_DIM) / M_TILE, 256, 0, stream>>>(xt, wk, table, bias, out);
}